// Linear_79963701117038
// MI455X (gfx1250) — compile-verified
//
#include <hip/hip_runtime.h>

// Equivariant linear (256x0e + 128x1o + 64x2e + 32x3o) on gfx1250.
// Per irrep block:  Y = (1/sqrt(mul)) * W @ X,  K = M = mul, N = BATCH*d.
// Wave-level 2x2 register blocking around V_WMMA_F32_16X16X4_F32:
//   4 independent accumulator chains, 1 memory op per WMMA, all loads
//   immediate-offset (zero inner-loop address VALU).

typedef __attribute__((ext_vector_type(2))) float v2f;
typedef __attribute__((ext_vector_type(4))) float v4f;
typedef __attribute__((ext_vector_type(8))) float v8f;

#define FEAT   1184
#define NBATCH 131072
#define BD     256

__device__ __forceinline__ v8f wmma4(v2f a, v2f b, v8f c) {
    return __builtin_amdgcn_wmma_f32_16x16x4_f32(
        false, a, false, b, (short)0, c, false, false);
}

// Store one 16x16 C/D fragment: VGPR r -> row vb+r (lanes 0-15) / vb+8+r
// (lanes 16-31, folded into vb by caller), column fixed per lane.
template<int D>
__device__ __forceinline__ void store_acc(float* obase, int vb, v8f acc,
                                          float scale) {
    if constexpr (D == 1) {
        v4f lo = {acc[0] * scale, acc[1] * scale, acc[2] * scale, acc[3] * scale};
        v4f hi = {acc[4] * scale, acc[5] * scale, acc[6] * scale, acc[7] * scale};
        *(v4f*)(obase + vb)     = lo;      // global_store_b128
        *(v4f*)(obase + vb + 4) = hi;
    } else {
        #pragma unroll
        for (int r = 0; r < 8; ++r)
            obase[(vb + r) * D] = acc[r] * scale;
    }
}

template<int MUL, int D, int NB, int XO, int WO>
__global__ __launch_bounds__(BD)
void eqlin_wmma(const float* __restrict__ x, const float* __restrict__ w,
                float* __restrict__ out)
{
    constexpr int SEG  = MUL * D;       // floats per batch row in this segment
    constexpr int NC   = NB * D;        // X-tile columns (multiple of 32)
    constexpr int XSTR = MUL + 2;       // even (b64-aligned), !=0 mod 64 banks
    constexpr int MT32 = MUL / 32;      // 32-row wave tiles over M
    constexpr int NT32 = NC / 32;       // 32-col wave tiles over N
    constexpr int WT   = MT32 * NT32;
    static_assert(NC % 32 == 0, "NC must be a multiple of 32");
    static_assert((NB * SEG / 4) % BD == 0, "X tile must stage evenly");

    __shared__ float sX[NC * XSTR];     // X^T tile: sX[c*XSTR + u]

    const int tid  = threadIdx.x;
    const int lane = tid & 31;
    const int wave = tid >> 5;
    const int bOrg = (int)blockIdx.x * NB;

    const float scale = (MUL == 256) ? 0.0625f
                      : (MUL == 128) ? 0.08838834764831845f
                      : (MUL == 64)  ? 0.125f
                                     : 0.17677669529663687f;

    // Warm the (L2-resident) weight segment: global_prefetch_b8 per 128B line.
    for (int p = tid * 32; p < MUL * MUL; p += BD * 32)
        __builtin_prefetch(w + WO + p, 0, 3);

    // ---- stage X tile: coalesced float4 global reads, transposed into LDS ----
    #pragma unroll
    for (int it = 0; it < (NB * SEG / 4) / BD; ++it) {
        const int e4 = (it * BD + tid) * 4;
        const int bl = e4 / SEG;
        const int o4 = e4 % SEG;
        v4f v = *(const v4f*)(x + (size_t)(bOrg + bl) * FEAT + XO + o4);
        #pragma unroll
        for (int j = 0; j < 4; ++j) {
            const int o = o4 + j;
            const int u = o / D;        // folds away for D==1
            const int i = o - u * D;
            sX[(bl * D + i) * XSTR + u] = v[j];
        }
    }

    __syncthreads();

    // A frag (16x4): lanes 0-15 K={k0,k0+1}, lanes 16-31 K={k0+2,k0+3}; M=lane&15.
    // B frag (4x16): same K split; N = lane&15.
    const int mrow  = lane & 15;
    const int khalf = (lane >> 4) * 2;

    for (int t = wave; t < WT; t += BD / 32) {   // wave-uniform: EXEC all ones
        const int m0 = (t % MT32) * 32;
        const int n0 = (t / MT32) * 32;

        const float* a0p = w + WO + (m0 + mrow) * MUL + khalf;   // rows m0..+15
        const float* a1p = a0p + 16 * MUL;                       // rows +16..+31
        const float* b0p = sX + (n0 + mrow) * XSTR + khalf;      // cols n0..+15
        const float* b1p = b0p + 16 * XSTR;                      // cols +16..+31

        v8f acc00 = {0.f,0.f,0.f,0.f,0.f,0.f,0.f,0.f};
        v8f acc01 = {0.f,0.f,0.f,0.f,0.f,0.f,0.f,0.f};
        v8f acc10 = {0.f,0.f,0.f,0.f,0.f,0.f,0.f,0.f};
        v8f acc11 = {0.f,0.f,0.f,0.f,0.f,0.f,0.f,0.f};

        #pragma unroll 4
        for (int k0 = 0; k0 < MUL; k0 += 4) {
            v2f a0 = *(const v2f*)(a0p + k0);    // global_load_b64, imm offset
            v2f a1 = *(const v2f*)(a1p + k0);
            v2f b0 = *(const v2f*)(b0p + k0);    // ds_load_b64, imm offset
            v2f b1 = *(const v2f*)(b1p + k0);
            acc00 = wmma4(a0, b0, acc00);        // 4 independent chains
            acc01 = wmma4(a0, b1, acc01);
            acc10 = wmma4(a1, b0, acc10);
            acc11 = wmma4(a1, b1, acc11);
        }

        // Columns of the two n-subtiles for this lane.
        const int c0 = n0 + mrow;
        const int c1 = c0 + 16;
        float* ob0 = out + (size_t)(bOrg + c0 / D) * FEAT + XO + (c0 % D);
        float* ob1 = out + (size_t)(bOrg + c1 / D) * FEAT + XO + (c1 % D);
        const int vb0 = m0 + (lane >> 4) * 8;    // rows of m-subtile 0
        const int vb1 = vb0 + 16;                // rows of m-subtile 1

        store_acc<D>(ob0, vb0, acc00, scale);
        store_acc<D>(ob1, vb0, acc01, scale);
        store_acc<D>(ob0, vb1, acc10, scale);
        store_acc<D>(ob1, vb1, acc11, scale);
    }
}

extern "C" void kernel_launch(void* const* d_in, const int* in_sizes, int n_in,
                              void* d_out, int out_size, void* d_ws, size_t ws_size,
                              hipStream_t stream) {
    (void)in_sizes; (void)n_in; (void)out_size; (void)d_ws; (void)ws_size;
    const float* x = (const float*)d_in[0];
    const float* w = (const float*)d_in[1];
    float* out = (float*)d_out;

    // <MUL, D, NB, XO, WO>   (full M per workgroup -> x read from HBM once)
    // 256x0e: 33KB LDS, 8 wave-tiles/WG, 64 WMMA-steps x4
    eqlin_wmma<256, 1, 32,   0,     0>
        <<<dim3(NBATCH / 32), BD, 0, stream>>>(x, w, out);
    // 128x1o: 50KB LDS, 12 wave-tiles/WG
    eqlin_wmma<128, 3, 32, 256, 65536>
        <<<dim3(NBATCH / 32), BD, 0, stream>>>(x, w, out);
    // 64x2e: 42KB LDS, 10 wave-tiles/WG
    eqlin_wmma< 64, 5, 32, 640, 81920>
        <<<dim3(NBATCH / 32), BD, 0, stream>>>(x, w, out);
    // 32x3o: 61KB LDS, 14 wave-tiles/WG
    eqlin_wmma< 32, 7, 64, 960, 86016>
        <<<dim3(NBATCH / 64), BD, 0, stream>>>(x, w, out);
}